// butterflyAutoEncode_62981400428623
// MI455X (gfx1250) — compile-verified
//
#include <hip/hip_runtime.h>

typedef __attribute__((ext_vector_type(2))) float v2f;
typedef __attribute__((ext_vector_type(8))) float v8f;
typedef __attribute__((ext_vector_type(4))) unsigned int u32x4;
typedef __attribute__((ext_vector_type(4))) int i32x4;
typedef __attribute__((ext_vector_type(8))) int i32x8;

#define D_DIM 1024
#define M_DIM 20
#define N_DIM 8192
#define K_SV  10

static __device__ __forceinline__ v8f wmma4(v2f a, v2f b, v8f c) {
  // V_WMMA_F32_16X16X4_F32: D = A(16x4,f32) * B(4x16,f32) + C(16x16,f32)
  return __builtin_amdgcn_wmma_f32_16x16x4_f32(
      /*neg_a=*/false, a, /*neg_b=*/false, b,
      /*c_mod=*/(short)0, c, /*reuse_a=*/false, /*reuse_b=*/false);
}

// ---------------------------------------------------------------------------
// TDM: DMA `n` contiguous floats from global memory to LDS offset 0.
// Descriptor per CDNA5 ISA §8.3/8.4: 2D tile n x 1, data_size=4B, type=2.
// Issued by the calling wave only (TDM ignores EXEC); completion via
// s_wait_tensorcnt 0.  Caller must ensure destination is at LDS base.
// ---------------------------------------------------------------------------
static __device__ __forceinline__ void tdm_load_floats_to_lds0(const float* gsrc, int n) {
#if __has_builtin(__builtin_amdgcn_tensor_load_to_lds)
  const unsigned long long ga = (unsigned long long)(size_t)gsrc;
  u32x4 g0;
  g0[0] = 1u;                                              // count=1 (valid user D#)
  g0[1] = 0u;                                              // lds_addr = 0
  g0[2] = (unsigned)(ga & 0xFFFFFFFFu);                    // global_addr[31:0]
  g0[3] = (unsigned)((ga >> 32) & 0x01FFFFFFu) | (2u << 30); // addr[56:32] | type=2
  i32x8 g1;
  g1[0] = (int)(2u << 16);                                 // data_size=2 (4 bytes)
  g1[1] = (int)(((unsigned)n & 0xFFFFu) << 16);            // tensor_dim0[15:0]
  g1[2] = (int)((((unsigned)n >> 16) & 0xFFFFu) | (1u << 16)); // dim0 hi | tensor_dim1=1
  g1[3] = (int)(((unsigned)n & 0xFFFFu) << 16);            // tile_dim0 = n
  g1[4] = 1;                                               // tile_dim1 = 1
  g1[5] = n;                                               // tensor_dim0_stride[31:0]
  g1[6] = 0;
  g1[7] = 0;
  i32x4 z4 = {0, 0, 0, 0};
#if defined(__clang_major__) && (__clang_major__ >= 23)
  i32x8 z8 = {0, 0, 0, 0, 0, 0, 0, 0};
  __builtin_amdgcn_tensor_load_to_lds(g0, g1, z4, z4, z8, 0);
#else
  __builtin_amdgcn_tensor_load_to_lds(g0, g1, z4, z4, 0);
#endif
#if __has_builtin(__builtin_amdgcn_s_wait_tensorcnt)
  __builtin_amdgcn_s_wait_tensorcnt(0);
#endif
#else
  (void)gsrc; (void)n;
#endif
}

// ---------------------------------------------------------------------------
// GEMM1: xx[20,8192] = W[20,1024] @ x[1024,8192].  1024 waves, 1 tile/wave.
// ---------------------------------------------------------------------------
__global__ void gemm1_Wx(const float* __restrict__ W, const float* __restrict__ x,
                         float* __restrict__ xx) {
  const int wave  = (blockIdx.x * blockDim.x + threadIdx.x) >> 5;
  const int lane  = threadIdx.x & 31;
  const int mTile = wave & 1;          // 2 m-tiles (rows 0..31, 20 valid)
  const int nTile = wave >> 1;         // 512 n-tiles
  const int row   = mTile * 16 + (lane & 15);
  const int col   = nTile * 16 + (lane & 15);
  const int khalf = (lane >> 4) * 2;
  const int rowc  = (row < M_DIM) ? row : 0;     // clamped address, no EXEC branch
  const float am  = (row < M_DIM) ? 1.f : 0.f;   // zero-mask for pad rows
  v8f acc = {0.f,0.f,0.f,0.f,0.f,0.f,0.f,0.f};
  for (int k0 = 0; k0 < D_DIM; k0 += 4) {
    const int kk = k0 + khalf;
    __builtin_prefetch(x + (size_t)(k0 + 8) * N_DIM + col, 0, 3);
    v2f a = *reinterpret_cast<const v2f*>(W + (size_t)rowc * D_DIM + kk);
    a.x *= am; a.y *= am;
    v2f b;
    b.x = x[(size_t)kk       * N_DIM + col];
    b.y = x[(size_t)(kk + 1) * N_DIM + col];
    acc = wmma4(a, b, acc);
  }
#pragma unroll
  for (int r = 0; r < 8; ++r) {
    const int mrow = mTile * 16 + r + ((lane >> 4) << 3);
    if (mrow < M_DIM) xx[(size_t)mrow * N_DIM + col] = acc[r];
  }
}

// ---------------------------------------------------------------------------
// GEMM2: AU[1024,20] = x[1024,8192] @ V2[8192,20].  128 waves.
// ---------------------------------------------------------------------------
__global__ void gemm2_xV2(const float* __restrict__ x, const float* __restrict__ V2,
                          float* __restrict__ AU) {
  const int wave  = (blockIdx.x * blockDim.x + threadIdx.x) >> 5;
  const int lane  = threadIdx.x & 31;
  const int nTile = wave & 1;          // 2 n-tiles (cols 0..31, 20 valid)
  const int mTile = wave >> 1;         // 64 m-tiles
  const int row   = mTile * 16 + (lane & 15);
  const int col   = nTile * 16 + (lane & 15);
  const int khalf = (lane >> 4) * 2;
  const int colc  = (col < M_DIM) ? col : 0;
  const float bm  = (col < M_DIM) ? 1.f : 0.f;
  v8f acc = {0.f,0.f,0.f,0.f,0.f,0.f,0.f,0.f};
  for (int k0 = 0; k0 < N_DIM; k0 += 4) {
    const int kk = k0 + khalf;
    v2f a = *reinterpret_cast<const v2f*>(x + (size_t)row * N_DIM + kk);
    v2f b;
    b.x = V2[(size_t)kk       * M_DIM + colc] * bm;
    b.y = V2[(size_t)(kk + 1) * M_DIM + colc] * bm;
    acc = wmma4(a, b, acc);
  }
  if (col < M_DIM) {
#pragma unroll
    for (int r = 0; r < 8; ++r) {
      const int mrow = mTile * 16 + r + ((lane >> 4) << 3);
      AU[(size_t)mrow * M_DIM + col] = acc[r];
    }
  }
}

// ---------------------------------------------------------------------------
// GEMM3: out[1024,8192] = T[1024,20] @ V2^T (V2 stored [8192,20]). 32768 waves.
// ---------------------------------------------------------------------------
__global__ void gemm3_TV2T(const float* __restrict__ T, const float* __restrict__ V2,
                           float* __restrict__ out) {
  const int wave  = (blockIdx.x * blockDim.x + threadIdx.x) >> 5;
  const int lane  = threadIdx.x & 31;
  const int nTile = wave & 511;        // 512 n-tiles
  const int mTile = wave >> 9;         // 64 m-tiles
  const int row   = mTile * 16 + (lane & 15);
  const int col   = nTile * 16 + (lane & 15);
  const int khalf = (lane >> 4) * 2;
  v8f acc = {0.f,0.f,0.f,0.f,0.f,0.f,0.f,0.f};
#pragma unroll
  for (int k0 = 0; k0 < M_DIM; k0 += 4) {   // K=20 -> exactly 5 steps
    const int kk = k0 + khalf;
    v2f a = *reinterpret_cast<const v2f*>(T  + (size_t)row * M_DIM + kk);
    v2f b = *reinterpret_cast<const v2f*>(V2 + (size_t)col * M_DIM + kk);
    acc = wmma4(a, b, acc);
  }
#pragma unroll
  for (int r = 0; r < 8; ++r) {
    const int mrow = mTile * 16 + r + ((lane >> 4) << 3);
    out[(size_t)mrow * N_DIM + col] = acc[r];
  }
}

// ---------------------------------------------------------------------------
// gram20: out[i,j] = sum_t A[i*rsA + t*es] * B[j*rsB + t*es]   (20x20 output)
// Used for G0 = xx xx^T, P = xx v1^T, H = AU^T AU.
// ---------------------------------------------------------------------------
__global__ void gram20(const float* __restrict__ A, const float* __restrict__ B,
                       int rsA, int rsB, int es, int L, float* __restrict__ out) {
  const int t = threadIdx.x;
  if (t >= 400) return;
  const int i = t / 20, j = t % 20;
  const float* pa = A + (size_t)i * rsA;
  const float* pb = B + (size_t)j * rsB;
  float acc = 0.f;
  for (int n = 0; n < L; ++n)
    acc += pa[(size_t)n * es] * pb[(size_t)n * es];
  out[i * 20 + j] = acc;
}

// ---------------------------------------------------------------------------
// SVD#1 in Gram space. A0 = xx [20,8192], G = A A^T [20,20], z = A x.
// z <- Gz / sqrt(z^T G z);  w_j = Gz/n;  G <- G - w w^T;
// c_j = (z - sum c_i (w_i . z)) / n;  z0_j = P[:,j] - sum w_i (c_i . P[:,j]).
// Output: C [20,20], with V2 = xx^T C.
// ---------------------------------------------------------------------------
__global__ void svd1(const float* __restrict__ G0, const float* __restrict__ P,
                     float* __restrict__ Cout) {
  __shared__ float zs[32], gzs[32], ps[32];
  __shared__ float Wv[20][20], Cm[20][20];
  const int i = threadIdx.x;
  const bool act = (i < 20);
  float Grow[20];
#pragma unroll
  for (int k = 0; k < 20; ++k) Grow[k] = act ? G0[i * 20 + k] : 0.f;

  for (int j = 0; j < 20; ++j) {
    const float pi = act ? P[i * 20 + j] : 0.f;
    ps[i] = pi;
    __syncthreads();
    float zi = pi;
    for (int jp = 0; jp < j; ++jp) {
      float d = 0.f;
      for (int k = 0; k < 20; ++k) d += Cm[jp][k] * ps[k];
      if (act) zi -= Wv[jp][i] * d;
    }
    zs[i] = act ? zi : 0.f;
    __syncthreads();

    for (int t = 0; t < 299; ++t) {            // 299 z-updates -> z_299
      float g = 0.f;
      for (int k = 0; k < 20; ++k) g += Grow[k] * zs[k];
      __syncthreads();
      gzs[i] = act ? g : 0.f;
      __syncthreads();
      float nsq = 0.f;
      for (int k = 0; k < 20; ++k) nsq += zs[k] * gzs[k];
      const float inv = rsqrtf(nsq);
      __syncthreads();
      zs[i] = act ? g * inv : 0.f;
      __syncthreads();
    }

    // final application: n_299, w_j, c_j
    float g = 0.f;
    for (int k = 0; k < 20; ++k) g += Grow[k] * zs[k];
    __syncthreads();
    gzs[i] = act ? g : 0.f;
    __syncthreads();
    float nsq = 0.f;
    for (int k = 0; k < 20; ++k) nsq += zs[k] * gzs[k];
    const float invn = 1.f / sqrtf(nsq);
    const float wi = g * invn;
    float ci = act ? zs[i] : 0.f;
    for (int jp = 0; jp < j; ++jp) {
      float d = 0.f;
      for (int k = 0; k < 20; ++k) d += Wv[jp][k] * zs[k];
      if (act) ci -= Cm[jp][i] * d;
    }
    ci *= invn;
    __syncthreads();
    if (act) {
      Wv[j][i] = wi;
      Cm[j][i] = ci;
      Cout[j * 20 + i] = ci;
    }
    __syncthreads();
    for (int k = 0; k < 20; ++k) Grow[k] -= wi * Wv[j][k];   // G <- G - w w^T
  }
}

// ---------------------------------------------------------------------------
// V2[q,j] = sum_k xx[k,q] * C[j,k].   C staged into LDS via TDM + fallback.
// ---------------------------------------------------------------------------
__global__ void v2_from_C(const float* __restrict__ xx, const float* __restrict__ C,
                          float* __restrict__ V2) {
  __shared__ float Cs[400];          // only __shared__ object -> LDS offset 0
  if (threadIdx.x < 32) tdm_load_floats_to_lds0(C, 400);
  __syncthreads();
  // Robustness fill (same values; TDM result already landed before the barrier)
  for (int t = threadIdx.x; t < 400; t += blockDim.x) Cs[t] = C[t];
  __syncthreads();
  const int q = blockIdx.x * blockDim.x + threadIdx.x;
  if (q >= N_DIM) return;
  float acc[20];
#pragma unroll
  for (int j = 0; j < 20; ++j) acc[j] = 0.f;
  for (int k = 0; k < 20; ++k) {
    const float xv = xx[(size_t)k * N_DIM + q];
#pragma unroll
    for (int j = 0; j < 20; ++j) acc[j] += xv * Cs[j * 20 + k];
  }
#pragma unroll
  for (int j = 0; j < 20; ++j) V2[(size_t)q * 20 + j] = acc[j];
}

// ---------------------------------------------------------------------------
// SVD#2 in 20-dim space. H = AU^T AU; x <- Hx/||Hx|| (300 iters);
// d_j = xf_j - sum d_i (xf_i . xf_j);  M += d_j xf_j^T;
// H <- H - (Hxf)xf^T - xf(Hxf)^T + (xf^T H xf) xf xf^T.
// Then (U3*S3) V3^T = AU @ M.
// ---------------------------------------------------------------------------
__global__ void svd2(const float* __restrict__ H, const float* __restrict__ vinit,
                     float* __restrict__ Mout) {
  __shared__ float xs[32], hxs[32];
  __shared__ float XF[K_SV][20], Dv[K_SV][20];
  const int i = threadIdx.x;
  const bool act = (i < 20);
  float Hrow[20], Mrow[20];
#pragma unroll
  for (int k = 0; k < 20; ++k) { Hrow[k] = act ? H[i * 20 + k] : 0.f; Mrow[k] = 0.f; }

  for (int j = 0; j < K_SV; ++j) {
    xs[i] = act ? vinit[j * 20 + i] : 0.f;
    __syncthreads();
    for (int t = 0; t < 300; ++t) {
      float hx = 0.f;
      for (int k = 0; k < 20; ++k) hx += Hrow[k] * xs[k];
      __syncthreads();
      hxs[i] = act ? hx : 0.f;
      __syncthreads();
      float nsq = 0.f;
      for (int k = 0; k < 20; ++k) nsq += hxs[k] * hxs[k];
      const float inv = rsqrtf(nsq);
      __syncthreads();
      xs[i] = act ? hx * inv : 0.f;
      __syncthreads();
    }
    // xf in xs (unit)
    float di = act ? xs[i] : 0.f;
    for (int jp = 0; jp < j; ++jp) {
      float dot = 0.f;
      for (int k = 0; k < 20; ++k) dot += XF[jp][k] * xs[k];
      if (act) di -= Dv[jp][i] * dot;
    }
    for (int k = 0; k < 20; ++k) Mrow[k] += di * xs[k];
    // H deflation
    float hx = 0.f;
    for (int k = 0; k < 20; ++k) hx += Hrow[k] * xs[k];
    __syncthreads();
    hxs[i] = act ? hx : 0.f;
    __syncthreads();
    float q = 0.f;
    for (int k = 0; k < 20; ++k) q += xs[k] * hxs[k];
    const float xi = act ? xs[i] : 0.f;
    for (int k = 0; k < 20; ++k)
      Hrow[k] -= hx * xs[k] + xi * hxs[k] - q * xi * xs[k];
    __syncthreads();
    if (act) { XF[j][i] = xs[i]; Dv[j][i] = di; }
    __syncthreads();
  }
  if (act)
    for (int k = 0; k < 20; ++k) Mout[i * 20 + k] = Mrow[k];
}

// ---------------------------------------------------------------------------
// T[1024,20] = AU[1024,20] @ M[20,20].   M staged into LDS via TDM + fallback.
// ---------------------------------------------------------------------------
__global__ void t_from_M(const float* __restrict__ AU, const float* __restrict__ M,
                         float* __restrict__ T) {
  __shared__ float Ms[400];          // only __shared__ object -> LDS offset 0
  if (threadIdx.x < 32) tdm_load_floats_to_lds0(M, 400);
  __syncthreads();
  for (int t = threadIdx.x; t < 400; t += blockDim.x) Ms[t] = M[t];
  __syncthreads();
  const int r = blockIdx.x * blockDim.x + threadIdx.x;
  if (r >= D_DIM) return;
  float a[20];
#pragma unroll
  for (int k = 0; k < 20; ++k) a[k] = AU[(size_t)r * 20 + k];
#pragma unroll
  for (int j = 0; j < 20; ++j) {
    float acc = 0.f;
#pragma unroll
    for (int k = 0; k < 20; ++k) acc += a[k] * Ms[k * 20 + j];
    T[(size_t)r * 20 + j] = acc;
  }
}

extern "C" void kernel_launch(void* const* d_in, const int* in_sizes, int n_in,
                              void* d_out, int out_size, void* d_ws, size_t ws_size,
                              hipStream_t stream) {
  (void)in_sizes; (void)n_in; (void)out_size; (void)ws_size;
  const float* x   = (const float*)d_in[0];   // [1024, 8192]
  const float* W   = (const float*)d_in[1];   // [20, 1024]
  const float* v1  = (const float*)d_in[2];   // [20, 8192]
  const float* v2i = (const float*)d_in[3];   // [10, 20]
  float* out = (float*)d_out;                 // [1024, 8192]

  float* ws = (float*)d_ws;
  float* xx = ws;                   // 20*8192
  float* V2 = xx + 20 * 8192;       // 8192*20
  float* AU = V2 + 8192 * 20;       // 1024*20
  float* T  = AU + 1024 * 20;       // 1024*20
  float* G0 = T  + 1024 * 20;       // 400
  float* P  = G0 + 400;             // 400
  float* C  = P  + 400;             // 400
  float* H  = C  + 400;             // 400
  float* M  = H  + 400;             // 400

  gemm1_Wx  <<<128, 256, 0, stream>>>(W, x, xx);
  gram20    <<<1, 512, 0, stream>>>(xx, xx, N_DIM, N_DIM, 1, N_DIM, G0); // G = xx xx^T
  gram20    <<<1, 512, 0, stream>>>(xx, v1, N_DIM, N_DIM, 1, N_DIM, P);  // P = xx v1^T
  svd1      <<<1, 32, 0, stream>>>(G0, P, C);
  v2_from_C <<<32, 256, 0, stream>>>(xx, C, V2);
  gemm2_xV2 <<<16, 256, 0, stream>>>(x, V2, AU);
  gram20    <<<1, 512, 0, stream>>>(AU, AU, 1, 1, M_DIM, D_DIM, H);      // H = AU^T AU
  svd2      <<<1, 32, 0, stream>>>(H, v2i, M);
  t_from_M  <<<4, 256, 0, stream>>>(AU, M, T);
  gemm3_TV2T<<<4096, 256, 0, stream>>>(T, V2, out);
}